// StochVolSimulator_40415642255701
// MI455X (gfx1250) — compile-verified
//
#include <hip/hip_runtime.h>

// ---------------------------------------------------------------------------
// StochVol simulator for MI455X (gfx1250, wave32).
// 3 kernels, single effective pass over HBM (~64 MiB write ≈ bandwidth floor
// at 23.3 TB/s). eps is regenerated from a counter-based RNG instead of being
// stored. Intra-wave 256-element prefix sum via 4 chained
// V_WMMA_F32_16X16X4_F32 (data x upper-triangular-ones = row prefix).
// Round-1 fixes: v_readlane instead of ds_bpermute for row-total broadcast;
// reciprocal-multiply instead of per-element IEEE division by tau.
// ---------------------------------------------------------------------------

typedef __attribute__((ext_vector_type(2))) float v2f;
typedef __attribute__((ext_vector_type(8))) float v8f;

#define N_TOTAL        16777217LL   // T + 1
#define BLOCK_THREADS  256
#define WAVES_PER_BLOCK 8
#define WAVE_ELEMS     256
#define BLOCK_ELEMS    2048
#define NBLOCKS        8193         // ceil(N_TOTAL / BLOCK_ELEMS)

#define STREAM_EPS   0x1001u
#define STREAM_NORM  0x2002u
#define STREAM_GNRM  0x3003u
#define STREAM_GACC  0x4004u
#define STREAM_GBST  0x5005u

// ----------------------------- RNG -----------------------------------------
__device__ __forceinline__ void tf2x32(uint32_t k0, uint32_t k1,
                                       uint32_t c0, uint32_t c1,
                                       uint32_t& o0, uint32_t& o1) {
  uint32_t ks2 = 0x1BD11BDAu ^ k0 ^ k1;
  uint32_t x0 = c0 + k0, x1 = c1 + k1;
#define TF_R(r) { x0 += x1; x1 = (x1 << (r)) | (x1 >> (32 - (r))); x1 ^= x0; }
  TF_R(13) TF_R(15) TF_R(26) TF_R(6)
  x0 += k1;  x1 += ks2 + 1u;
  TF_R(17) TF_R(29) TF_R(16) TF_R(24)
  x0 += ks2; x1 += k0 + 2u;
  TF_R(13) TF_R(15) TF_R(26) TF_R(6)
  x0 += k0;  x1 += k1 + 3u;
  TF_R(17) TF_R(29) TF_R(16) TF_R(24)
  x0 += k1;  x1 += ks2 + 4u;
  TF_R(13) TF_R(15) TF_R(26) TF_R(6)
  x0 += ks2; x1 += k0 + 5u;
#undef TF_R
  o0 = x0; o1 = x1;
}

__device__ __forceinline__ float u01(uint32_t x) {
  // (0,1) open interval
  return (float)(x >> 8) * 5.9604645e-08f + 2.9802322e-08f;
}

__device__ __forceinline__ float rng_normal(long long idx, uint32_t stream) {
  uint32_t a, b;
  tf2x32(0x9E3779B9u ^ stream, 0x0000002Au,
         (uint32_t)idx, (uint32_t)(idx >> 32) ^ stream, a, b);
  const float u1 = u01(a), u2 = u01(b);
  return sqrtf(-2.0f * __logf(u1)) * __cosf(6.28318530718f * u2);
}

__device__ __forceinline__ float rng_uniform(long long idx, uint32_t stream) {
  uint32_t a, b;
  tf2x32(0x9E3779B9u ^ stream, 0x0000002Au,
         (uint32_t)idx, (uint32_t)(idx >> 32) ^ stream, a, b);
  return u01(a);
}

// Marsaglia-Tsang gamma; alpha >= 0.25 (nu clamped to >= 0.5).
__device__ __noinline__ float rng_gamma(long long idx, float alpha) {
  float boost = 1.0f;
  float a = alpha;
  if (a < 1.0f) {
    boost = __powf(rng_uniform(idx, STREAM_GBST), 1.0f / alpha);
    a += 1.0f;
  }
  const float d = a - (1.0f / 3.0f);
  const float c = rsqrtf(9.0f * d);
  float res = d;                       // deterministic fallback
#pragma unroll 1
  for (uint32_t t = 0; t < 6; ++t) {
    const float x = rng_normal(idx, STREAM_GNRM + (t << 16));
    float v = 1.0f + c * x;
    if (v <= 0.0f) continue;
    v = v * v * v;
    const float u  = rng_uniform(idx, STREAM_GACC + (t << 16));
    const float x2 = x * x;
    if (u < 1.0f - 0.0331f * x2 * x2 ||
        __logf(u) < 0.5f * x2 + d * (1.0f - v + __logf(v))) {
      res = d * v;
      break;
    }
  }
  return res * boost;
}

// Wave-uniform broadcast from a compile-time-constant lane: v_readlane_b32.
__device__ __forceinline__ float readlane_f32(float v, int lane) {
  return __int_as_float(__builtin_amdgcn_readlane(__float_as_int(v), lane));
}

// ------------------- Kernel 1: per-block eps sums ---------------------------
__global__ void __launch_bounds__(BLOCK_THREADS)
sv_blocksums(const float* __restrict__ theta, float* __restrict__ partials) {
  const int b = blockIdx.x;
  const int tid = threadIdx.x;
  const float tau = fmaxf(10.0f, theta[1]);
  const float inv_tau = 1.0f / tau;          // hoisted: one div per thread
  const long long base = (long long)b * BLOCK_ELEMS + (long long)tid * 8;
  float s = 0.0f;
#pragma unroll
  for (int e = 0; e < 8; ++e) {
    const long long idx = base + e;
    s += (idx < N_TOTAL) ? (rng_normal(idx, STREAM_EPS) * inv_tau) : 0.0f;
  }
#pragma unroll
  for (int o = 16; o >= 1; o >>= 1) s += __shfl_xor(s, o, 32);
  __shared__ float wred[WAVES_PER_BLOCK];
  if ((tid & 31) == 0) wred[tid >> 5] = s;
  __syncthreads();
  if (tid == 0) {
    float tot = 0.0f;
#pragma unroll
    for (int w = 0; w < WAVES_PER_BLOCK; ++w) tot += wred[w];
    partials[b] = tot;
  }
}

// ------------------- Kernel 2: scan the 8193 partials -----------------------
__global__ void __launch_bounds__(1024)
sv_scanpartials(const float* __restrict__ partials, float* __restrict__ offsets) {
  const int t = threadIdx.x;              // 1024 threads, 9 each covers 8193
  float loc_ex[9];
  float sum = 0.0f;
#pragma unroll
  for (int e = 0; e < 9; ++e) {
    const int g = t * 9 + e;
    const float v = (g < NBLOCKS) ? partials[g] : 0.0f;
    loc_ex[e] = sum;                      // exclusive within this thread
    sum += v;
  }
  __shared__ float ls[1024];
  ls[t] = sum;
  __syncthreads();
  for (int off = 1; off < 1024; off <<= 1) {   // Hillis-Steele inclusive
    const float x = (t >= off) ? ls[t - off] : 0.0f;
    __syncthreads();
    ls[t] += x;
    __syncthreads();
  }
  const float excl = (t > 0) ? ls[t - 1] : 0.0f;
#pragma unroll
  for (int e = 0; e < 9; ++e) {
    const int g = t * 9 + e;
    if (g < NBLOCKS) offsets[g] = excl + loc_ex[e];   // exclusive block offset
  }
}

// ------------------- Kernel 3: WMMA scan + StudentT output ------------------
__global__ void __launch_bounds__(BLOCK_THREADS)
sv_final(const float* __restrict__ theta,
         const float* __restrict__ blockoff,
         float* __restrict__ out) {
  const int b    = blockIdx.x;
  const int tid  = threadIdx.x;
  const int wave = tid >> 5;
  const int lane = tid & 31;
  const int half = lane >> 4;
  const int N    = lane & 15;   // A-operand row M == N for this lane; C col N
  const float nu  = fmaxf(0.5f,  theta[0]);
  const float tau = fmaxf(10.0f, theta[1]);
  const float inv_tau = 1.0f / tau;          // hoisted: one div per thread
  const long long g0 = (long long)b * BLOCK_ELEMS + (long long)wave * WAVE_ELEMS;

  // eps in the WMMA A-operand layout: lane l holds row M = l%16;
  // VGPR0/VGPR1 carry K = 4j + 2*half, 4j + 2*half + 1 for chunk j.
  float a[8];
#pragma unroll
  for (int j = 0; j < 4; ++j) {
    const int k0 = 4 * j + 2 * half;
    const long long i0 = g0 + (long long)N * 16 + k0;
    a[2 * j]     = (i0     < N_TOTAL) ? (rng_normal(i0,     STREAM_EPS) * inv_tau) : 0.0f;
    a[2 * j + 1] = (i0 + 1 < N_TOTAL) ? (rng_normal(i0 + 1, STREAM_EPS) * inv_tau) : 0.0f;
  }

  // D = X * U (U = inclusive upper-triangular ones) => row-wise prefix sums.
  // Chained over K in 4 steps of 16x16x4 f32 WMMA.
  v8f c = {0.0f, 0.0f, 0.0f, 0.0f, 0.0f, 0.0f, 0.0f, 0.0f};
#pragma unroll
  for (int j = 0; j < 4; ++j) {
    v2f av; av.x = a[2 * j]; av.y = a[2 * j + 1];
    v2f bv;
    bv.x = ((4 * j + 2 * half)     <= N) ? 1.0f : 0.0f;
    bv.y = ((4 * j + 2 * half + 1) <= N) ? 1.0f : 0.0f;
    c = __builtin_amdgcn_wmma_f32_16x16x4_f32(false, av, false, bv,
                                              (short)0, c, false, false);
  }

  float cc[8];
#pragma unroll
  for (int v = 0; v < 8; ++v) cc[v] = c[v];

  // Row totals t[i] = D[i][15] live at lane 15 (rows 0..7) / lane 31
  // (rows 8..15), vgpr i%8. Broadcast with v_readlane (no LDS, no DScnt),
  // then exclusive-scan the 16 wave-uniform values.
  float ex[16];
  float run = 0.0f;
#pragma unroll
  for (int i = 0; i < 8; ++i) {
    ex[i] = run;
    run += readlane_f32(cc[i], 15);
  }
#pragma unroll
  for (int i = 8; i < 16; ++i) {
    ex[i] = run;
    run += readlane_f32(cc[i - 8], 31);
  }
  const float waveTotal = run;

  __shared__ float wsum[WAVES_PER_BLOCK];
  if (lane == 0) wsum[wave] = waveTotal;
  __syncthreads();
  float woff = 0.0f;
#pragma unroll
  for (int w = 0; w < WAVES_PER_BLOCK; ++w) woff += (w < wave) ? wsum[w] : 0.0f;

  const float base   = blockoff[b] + woff;
  const float inv_nu = 1.0f / nu;
  const float alpha  = 0.5f * nu;

  // C layout: lane l, vgpr v -> row Mi = v + 8*half, col N = l%16.
#pragma unroll
  for (int v = 0; v < 8; ++v) {
    const int Mi = v + 8 * half;
    const float exMi = half ? ex[v + 8] : ex[v];
    const long long idx = g0 + (long long)Mi * 16 + N;
    if (idx < N_TOTAL) {
      const float s    = base + exMi + cc[v];   // global inclusive cumsum
      const float vol  = __expf(s);
      const float z    = rng_normal(idx, STREAM_NORM);
      const float chi2 = 2.0f * rng_gamma(idx, alpha);
      out[idx] = z * rsqrtf(chi2 * inv_nu) * vol;
    }
  }
}

// ---------------------------------------------------------------------------
extern "C" void kernel_launch(void* const* d_in, const int* in_sizes, int n_in,
                              void* d_out, int out_size, void* d_ws, size_t ws_size,
                              hipStream_t stream) {
  (void)in_sizes; (void)n_in; (void)out_size; (void)ws_size;
  const float* theta = (const float*)d_in[0];
  float* out      = (float*)d_out;
  float* partials = (float*)d_ws;           // NBLOCKS floats
  float* offsets  = partials + NBLOCKS;     // NBLOCKS floats (~64 KiB total)

  sv_blocksums<<<NBLOCKS, BLOCK_THREADS, 0, stream>>>(theta, partials);
  sv_scanpartials<<<1, 1024, 0, stream>>>(partials, offsets);
  sv_final<<<NBLOCKS, BLOCK_THREADS, 0, stream>>>(theta, offsets, out);
}